// DM_20942260535952
// MI455X (gfx1250) — compile-verified
//
#include <hip/hip_runtime.h>

typedef __attribute__((ext_vector_type(2))) float v2f;
typedef __attribute__((ext_vector_type(8))) float v8f;

#define VEC_DIM   300
#define NUM_WORDS 100000
#define NUM_CTX   10
#define NUM_NOISE 26
#define A_ROWS    32   // 26 noise rows padded to 2 WMMA M-tiles

__global__ __launch_bounds__(32) void pvdm_score_kernel(
    const int*   __restrict__ context_ids,   // (B,10)
    const int*   __restrict__ doc_ids,       // (B,)
    const int*   __restrict__ noise_ids,     // (B,26)
    const float* __restrict__ D,             // (500000,300)
    const float* __restrict__ W,             // (100000,300)
    const float* __restrict__ O,             // (300,100000)
    float*       __restrict__ out)           // (B,26)
{
    __shared__ float sx[304];                 // x[b], padded
    __shared__ float sA[A_ROWS * VEC_DIM];    // gathered O columns, row n = noise n
    __shared__ int   swords[NUM_NOISE];

    const int b    = blockIdx.x;
    const int lane = threadIdx.x;

    // ---- stage noise word ids (uniform per block) ----
    if (lane < NUM_NOISE) swords[lane] = noise_ids[b * NUM_NOISE + lane];

    // ---- phase 1: x = D[doc] + sum_c W[ctx[c]] ----
    const long doc = doc_ids[b];
    int ctx[NUM_CTX];
    #pragma unroll
    for (int c = 0; c < NUM_CTX; ++c) ctx[c] = context_ids[b * NUM_CTX + c];

    for (int d = lane; d < VEC_DIM; d += 32) {
        float acc = D[doc * VEC_DIM + d];
        #pragma unroll
        for (int c = 0; c < NUM_CTX; ++c)
            acc += W[(long)ctx[c] * VEC_DIM + d];
        sx[d] = acc;
    }
    __syncthreads();   // single-wave WG: collapses to dscnt/loadcnt waits

    // ---- phase 2: gather 26 O-columns (stride NUM_WORDS) into LDS tile ----
    for (int idx = lane; idx < NUM_NOISE * VEC_DIM; idx += 32) {
        const int n = idx / VEC_DIM;
        const int d = idx - n * VEC_DIM;
        sA[n * VEC_DIM + d] = O[(long)d * NUM_WORDS + swords[n]];
    }
    // zero-pad rows 26..31 so the second M-tile is well-defined
    for (int idx = lane; idx < (A_ROWS - NUM_NOISE) * VEC_DIM; idx += 32)
        sA[NUM_NOISE * VEC_DIM + idx] = 0.0f;
    __syncthreads();

    // ---- phase 3: scores = A(32x300) x B(300x16), B columns = x broadcast ----
    // Fragment addressing per CDNA5 ISA 7.12.2 (32-bit A 16x4 / B 4x16 layouts):
    const int mrow = lane & 15;            // A row for this lane
    const int koff = (lane >> 4) << 1;     // K sub-offset: 0 (lanes 0-15) / 2 (lanes 16-31)

    v8f acc0 = {};   // noise rows 0..15
    v8f acc1 = {};   // noise rows 16..31
    for (int k = 0; k < VEC_DIM; k += 4) {
        const v2f bv = *(const v2f*)&sx[k + koff];                          // broadcast x
        const v2f a0 = *(const v2f*)&sA[ mrow       * VEC_DIM + k + koff];
        const v2f a1 = *(const v2f*)&sA[(mrow + 16) * VEC_DIM + k + koff];
        acc0 = __builtin_amdgcn_wmma_f32_16x16x4_f32(false, a0, false, bv,
                                                     (short)0, acc0, false, false);
        acc1 = __builtin_amdgcn_wmma_f32_16x16x4_f32(false, a1, false, bv,
                                                     (short)0, acc1, false, false);
    }

    // ---- write out: lane l, VGPR r holds D[M = r + 8*(l>=16), N = l%16];
    // every N column is identical (B columns identical), so any lane is valid.
    const int half8 = (lane >> 4) << 3;    // 0 or 8
    const int sub   = lane & 15;
    if (sub == 0) {
        #pragma unroll
        for (int r = 0; r < 8; ++r)
            out[(long)b * NUM_NOISE + half8 + r] = acc0[r];
    } else if (sub == 1) {
        #pragma unroll
        for (int r = 0; r < 8; ++r) {
            const int n = 16 + half8 + r;
            if (n < NUM_NOISE)
                out[(long)b * NUM_NOISE + n] = acc1[r];
        }
    }
}

extern "C" void kernel_launch(void* const* d_in, const int* in_sizes, int n_in,
                              void* d_out, int out_size, void* d_ws, size_t ws_size,
                              hipStream_t stream) {
    const int*   context_ids = (const int*)  d_in[0];
    const int*   doc_ids     = (const int*)  d_in[1];
    const int*   noise_ids   = (const int*)  d_in[2];
    const float* Dm          = (const float*)d_in[3];
    const float* Wm          = (const float*)d_in[4];
    const float* Om          = (const float*)d_in[5];
    float*       out         = (float*)d_out;

    const int batch = in_sizes[1];   // doc_ids element count == BATCH
    pvdm_score_kernel<<<batch, 32, 0, stream>>>(context_ids, doc_ids, noise_ids,
                                                Dm, Wm, Om, out);
}